// DeepFeatureLoss_9474697855457
// MI455X (gfx1250) — compile-verified
//
#include <hip/hip_runtime.h>

#define BATCH 2
#define NPTS  4096
#define DFEA  32
#define TILE  16
#define NCOLT (NPTS / TILE)            // 256 column tiles
#define WAVES 8
#define TPB   (WAVES * 32)             // 256 threads, 8 wave32s
#define TILES_PER_WAVE (NCOLT / WAVES) // 32
#define INV_SIGMA 200.0f               // 1 / 0.005

typedef __attribute__((ext_vector_type(2))) float v2f;
typedef __attribute__((ext_vector_type(8))) float v8f;

// ---------------------------------------------------------------------------
// Pass 1: per-row norms (points scaled by 1/sigma) and per-row reg sums.
// ---------------------------------------------------------------------------
__global__ void prep_kernel(const float* __restrict__ points,
                            const float* __restrict__ f1,
                            const float* __restrict__ f2,
                            float* __restrict__ sp,
                            float* __restrict__ sf1,
                            float* __restrict__ sf2,
                            float* __restrict__ regrow) {
  int i = blockIdx.x * blockDim.x + threadIdx.x;
  if (i >= BATCH * NPTS) return;
  const float* pr = points + (size_t)i * 3;
  float px = pr[0] * INV_SIGMA, py = pr[1] * INV_SIGMA, pz = pr[2] * INV_SIGMA;
  sp[i] = px * px + py * py + pz * pz;
  const float* r1 = f1 + (size_t)i * DFEA;
  const float* r2 = f2 + (size_t)i * DFEA;
  float a1 = 0.f, a2 = 0.f, reg = 0.f;
#pragma unroll
  for (int c = 0; c < DFEA; ++c) {
    float v1 = r1[c], v2 = r2[c];
    a1 += v1 * v1;
    a2 += v2 * v2;
    if (c >= 3) reg += v1 * v1 + v2 * v2;
  }
  sf1[i] = a1;
  sf2[i] = a2;
  regrow[i] = reg;
}

// ---------------------------------------------------------------------------
// Pass 2: fused dual-Gram + double-softmax cross entropy.
// One workgroup per 16-row tile; 8 waves split the 256 column tiles.
// Feature Gram: 8x chained V_WMMA_F32_16X16X4_F32 (K=32).
// Point Gram:   1x V_WMMA_F32_16X16X4_F32 (K=3 padded with 0).
//
// Point-side softmax uses a FIXED shift of 0: d_p <= 0 and d_p[i,i] == 0,
// so the row max is exactly 0 -> no online max/rescale needed.
// Feature-side logsumexp keeps the online-max update (max is data dependent).
// ---------------------------------------------------------------------------
__global__ __launch_bounds__(TPB)
void corres_kernel(const float* __restrict__ points,
                   const float* __restrict__ f1,
                   const float* __restrict__ f2,
                   const float* __restrict__ w,
                   const float* __restrict__ sp,
                   const float* __restrict__ sf1,
                   const float* __restrict__ sf2,
                   float* __restrict__ cepart) {
  __shared__ float lds[TILE * (WAVES * TILE) * 4]; // 16 rows x 128 slots x 4 stats = 32KB
  __shared__ float red[TILE];

  const int wg   = blockIdx.x;
  const int b    = wg / (NPTS / TILE);
  const int it   = wg % (NPTS / TILE);
  const int i0   = it * TILE;
  const int tid  = threadIdx.x;
  const int wave = tid >> 5;
  const int lane = tid & 31;
  const int lg   = lane >> 4;  // lane group (K / row-half select)
  const int lm   = lane & 15;  // lane within group (M for A, N for B)
  const int base = b * NPTS;

  // ---- A fragments (invariant over the j loop) ----
  // A[M=lm, K=2*lg+v] for K-step ks: f1[row, 4*ks + 2*lg + v]
  const float* f1row = f1 + (size_t)(base + i0 + lm) * DFEA + 2 * lg;
  v2f af[8];
#pragma unroll
  for (int ks = 0; ks < 8; ++ks) af[ks] = *(const v2f*)(f1row + ks * 4);

  const float* prow = points + (size_t)(base + i0 + lm) * 3;
  v2f ap;
  ap.x = prow[2 * lg] * INV_SIGMA;
  ap.y = (lg == 0) ? (prow[1] * INV_SIGMA) : 0.0f; // pad K=3 with zero

  // per-lane row constants: C VGPR r holds row (r + 8*lg)
  float spi[8], sfi[8];
#pragma unroll
  for (int r = 0; r < 8; ++r) {
    int row = base + i0 + r + 8 * lg;
    spi[r] = sp[row];
    sfi[r] = sf1[row];
  }

  // streaming state (per lane, 8 rows, one column slice)
  float z_p[8], t_s[8], m_f[8], z_f[8];
#pragma unroll
  for (int r = 0; r < 8; ++r) {
    z_p[r] = 0.f; t_s[r] = 0.f;
    m_f[r] = -__builtin_inff(); z_f[r] = 0.f;
  }

  for (int t = 0; t < TILES_PER_WAVE; ++t) {
    const int jt   = wave * TILES_PER_WAVE + t;
    const int j0   = jt * TILE;
    const int jrow = base + j0 + lm;

    // B fragments: B[K=2*lg+v, N=lm] = f2[j0+lm, 4*ks + 2*lg + v]
    const float* f2row = f2 + (size_t)jrow * DFEA + 2 * lg;
    v2f bfrag[8];
#pragma unroll
    for (int ks = 0; ks < 8; ++ks) bfrag[ks] = *(const v2f*)(f2row + ks * 4);

    const float* qrow = points + (size_t)jrow * 3;
    v2f bp;
    bp.x = qrow[2 * lg] * INV_SIGMA;
    bp.y = (lg == 0) ? (qrow[1] * INV_SIGMA) : 0.0f;

    const float sp_j  = sp[jrow];
    const float sf2_j = sf2[jrow];

    v8f cf = {};
#pragma unroll
    for (int ks = 0; ks < 8; ++ks)
      cf = __builtin_amdgcn_wmma_f32_16x16x4_f32(false, af[ks], false, bfrag[ks],
                                                 (short)0, cf, false, false);
    v8f cp = {};
    cp = __builtin_amdgcn_wmma_f32_16x16x4_f32(false, ap, false, bp,
                                               (short)0, cp, false, false);

#pragma unroll
    for (int r = 0; r < 8; ++r) {
      float dp = 2.f * cp[r] - spi[r] - sp_j; // <= 0, row max is exactly 0
      float df = 2.f * cf[r] - sfi[r] - sf2_j;
      // point side: fixed shift 0 -> single exp, plain accumulation
      float e = __expf(dp);
      z_p[r] += e;
      t_s[r] = fmaf(e, df, t_s[r]);
      // feature side: online logsumexp
      float mf = fmaxf(m_f[r], df);
      float sf = __expf(m_f[r] - mf);
      float ef = __expf(df - mf);
      z_f[r] = z_f[r] * sf + ef;
      m_f[r] = mf;
    }
  }

  // ---- combine partial stats across lanes and waves via LDS ----
  const int slot = wave * TILE + lm;
#pragma unroll
  for (int r = 0; r < 8; ++r) {
    int row = r + 8 * lg;
    float* dst = &lds[(row * (WAVES * TILE) + slot) * 4];
    dst[0] = z_p[r]; dst[1] = t_s[r];
    dst[2] = m_f[r]; dst[3] = z_f[r];
  }
  __syncthreads();

  if (tid < TILE) {
    float Z = 0.f, T = 0.f;
    float Mf = -__builtin_inff(), Zf = 0.f;
    const float* src = &lds[tid * (WAVES * TILE) * 4];
    for (int s = 0; s < WAVES * TILE; ++s) {
      Z += src[s * 4 + 0];
      T += src[s * 4 + 1];
      float mf1 = src[s * 4 + 2], zf1 = src[s * 4 + 3];
      float mfn = fmaxf(Mf, mf1);
      Zf = Zf * __expf(Mf - mfn) + zf1 * __expf(mf1 - mfn);
      Mf = mfn;
    }
    float Lf = Mf + __logf(Zf);
    float ce = Lf - T / Z; // ce_i = L_f - sum_j softmax_p * d_f
    red[tid] = ce * w[base + i0 + tid];
  }
  __syncthreads();
  if (tid == 0) {
    float acc = 0.f;
#pragma unroll
    for (int r = 0; r < TILE; ++r) acc += red[r];
    cepart[wg] = acc;
  }
}

// ---------------------------------------------------------------------------
// Pass 3: deterministic fixed-order reduction to the 4 outputs.
// ---------------------------------------------------------------------------
__global__ void reduce_kernel(const float* __restrict__ cepart,
                              const float* __restrict__ regrow,
                              float* __restrict__ out) {
  __shared__ float s[256];
  int tid = threadIdx.x;
  for (int b = 0; b < BATCH; ++b) {
    s[tid] = cepart[b * (NPTS / TILE) + tid];
    __syncthreads();
    for (int off = 128; off > 0; off >>= 1) {
      if (tid < off) s[tid] += s[tid + off];
      __syncthreads();
    }
    if (tid == 0) out[b] = s[0];
    __syncthreads();

    float a = 0.f;
    for (int k = tid; k < NPTS; k += 256) a += regrow[b * NPTS + k];
    s[tid] = a;
    __syncthreads();
    for (int off = 128; off > 0; off >>= 1) {
      if (tid < off) s[tid] += s[tid + off];
      __syncthreads();
    }
    if (tid == 0) out[2 + b] = s[0] / (29.0f * NPTS);
    __syncthreads();
  }
}

extern "C" void kernel_launch(void* const* d_in, const int* in_sizes, int n_in,
                              void* d_out, int out_size, void* d_ws, size_t ws_size,
                              hipStream_t stream) {
  const float* points = (const float*)d_in[0];
  const float* f1     = (const float*)d_in[1];
  const float* f2     = (const float*)d_in[2];
  const float* w      = (const float*)d_in[3];

  float* ws     = (float*)d_ws;
  float* sp     = ws;                       // [B*N]
  float* sf1    = ws + 1 * BATCH * NPTS;    // [B*N]
  float* sf2    = ws + 2 * BATCH * NPTS;    // [B*N]
  float* regrow = ws + 3 * BATCH * NPTS;    // [B*N]
  float* cepart = ws + 4 * BATCH * NPTS;    // [B * N/16] = 512

  prep_kernel<<<(BATCH * NPTS + 255) / 256, 256, 0, stream>>>(
      points, f1, f2, sp, sf1, sf2, regrow);
  corres_kernel<<<BATCH * (NPTS / TILE), TPB, 0, stream>>>(
      points, f1, f2, w, sp, sf1, sf2, cepart);
  reduce_kernel<<<1, 256, 0, stream>>>(cepart, regrow, (float*)d_out);
}